// GAT_10273561772449
// MI455X (gfx1250) — compile-verified
//
#include <hip/hip_runtime.h>
#include <math.h>

typedef __attribute__((ext_vector_type(2))) float v2f;
typedef __attribute__((ext_vector_type(8))) float v8f;

// ---------------------------------------------------------------- utilities
__global__ void fill_kernel(float* __restrict__ p, float v, long n) {
  long i = (long)blockIdx.x * blockDim.x + threadIdx.x;
  if (i < n) p[i] = v;
}

__device__ __forceinline__ void atomicMaxF(float* addr, float v) {
  // ordered-int trick; addr initialized to -inf
  if (v >= 0.0f) atomicMax((int*)addr, __float_as_int(v));
  else           atomicMin((unsigned int*)addr, __float_as_uint(v));
}

// ------------------------------------------------- WMMA GEMM, K=128 fixed
// C[M,NCOL] = A[M,128] * B[128,NCOL].  grid.x = ceil(M/16),
// blockDim = 32*(NCOL/16).  Each wave owns one 16x16 tile; K=128 is fully
// unrolled into 32 V_WMMA_F32_16X16X4_F32 spread over 4 independent
// accumulator chains.  A slab (16x128, 8KB) staged in LDS via float4.
template <int NCOL>
__global__ void __launch_bounds__(256) wmma_gemm_f32_k128(
    const float* __restrict__ A, const float* __restrict__ B,
    float* __restrict__ C, int M) {
  __shared__ float As[16 * 128];
  const int lane = threadIdx.x & 31;
  const int wave = threadIdx.x >> 5;
  const int row0 = blockIdx.x * 16;

  if (row0 + 16 <= M) {                       // fast path: full slab, b128
    const float4* A4  = (const float4*)(A + (long)row0 * 128);
    float4*       As4 = (float4*)As;
    for (int i = threadIdx.x; i < 16 * 128 / 4; i += blockDim.x) As4[i] = A4[i];
  } else {                                    // tail guard (unused for N=50000)
    for (int i = threadIdx.x; i < 16 * 128; i += blockDim.x) {
      int r = row0 + (i >> 7);
      if (r >= M) r = M - 1;
      As[i] = A[(long)r * 128 + (i & 127)];
    }
  }
  __syncthreads();

  const int col0  = wave * 16;
  const int mrow  = lane & 15;                // A row / B,C col within tile
  const int khalf = (lane >> 4) << 1;         // lanes 0-15: K+0/1, 16-31: K+2/3
  const float* Ap = As + mrow * 128 + khalf;
  const float* Bp = B + (long)khalf * NCOL + col0 + mrow;

  v8f acc0 = {}, acc1 = {}, acc2 = {}, acc3 = {};
#pragma unroll
  for (int k = 0; k < 128; k += 16) {
    v2f a0, a1, a2, a3, b0, b1, b2, b3;
    a0.x = Ap[k + 0];  a0.y = Ap[k + 1];
    a1.x = Ap[k + 4];  a1.y = Ap[k + 5];
    a2.x = Ap[k + 8];  a2.y = Ap[k + 9];
    a3.x = Ap[k + 12]; a3.y = Ap[k + 13];
    b0.x = Bp[(long)(k + 0)  * NCOL]; b0.y = Bp[(long)(k + 1)  * NCOL];
    b1.x = Bp[(long)(k + 4)  * NCOL]; b1.y = Bp[(long)(k + 5)  * NCOL];
    b2.x = Bp[(long)(k + 8)  * NCOL]; b2.y = Bp[(long)(k + 9)  * NCOL];
    b3.x = Bp[(long)(k + 12) * NCOL]; b3.y = Bp[(long)(k + 13) * NCOL];
    acc0 = __builtin_amdgcn_wmma_f32_16x16x4_f32(false, a0, false, b0, (short)0, acc0, false, false);
    acc1 = __builtin_amdgcn_wmma_f32_16x16x4_f32(false, a1, false, b1, (short)0, acc1, false, false);
    acc2 = __builtin_amdgcn_wmma_f32_16x16x4_f32(false, a2, false, b2, (short)0, acc2, false, false);
    acc3 = __builtin_amdgcn_wmma_f32_16x16x4_f32(false, a3, false, b3, (short)0, acc3, false, false);
  }
  v8f acc;
#pragma unroll
  for (int i = 0; i < 8; ++i) acc[i] = (acc0[i] + acc1[i]) + (acc2[i] + acc3[i]);

  // C/D layout (ISA 7.12.2): VGPR i -> row i (lanes 0-15) / row 8+i (16-31)
  const int rbase = row0 + ((lane >> 4) << 3);
  float* Cp = C + (long)rbase * NCOL + col0 + mrow;
  if (row0 + 16 <= M) {
#pragma unroll
    for (int i = 0; i < 8; ++i) Cp[(long)i * NCOL] = acc[i];
  } else {
#pragma unroll
    for (int i = 0; i < 8; ++i)
      if (rbase + i < M) Cp[(long)i * NCOL] = acc[i];
  }
}

// ------------------------------------------------- attention dot products
// el[n,h] = sum_d feat[n,h,d]*al[h,d];  er likewise.  one thread per (n,h).
template <int H, int D>
__global__ void attn_dots(const float* __restrict__ feat,
                          const float* __restrict__ al,
                          const float* __restrict__ ar,
                          float* __restrict__ el, float* __restrict__ er,
                          long NH) {
  long i = (long)blockIdx.x * blockDim.x + threadIdx.x;
  if (i >= NH) return;
  const int h = (int)(i & (H - 1));           // H is a power of two
  const float4* f  = (const float4*)(feat + i * D);
  const float4* l4 = (const float4*)(al + h * D);
  const float4* r4 = (const float4*)(ar + h * D);
  float sl = 0.f, sr = 0.f;
#pragma unroll
  for (int d = 0; d < D / 4; ++d) {
    float4 v = f[d], lv = l4[d], rv = r4[d];
    sl += v.x * lv.x + v.y * lv.y + v.z * lv.z + v.w * lv.w;
    sr += v.x * rv.x + v.y * rv.y + v.z * rv.z + v.w * rv.w;
  }
  el[i] = sl; er[i] = sr;
}

// ------------------------------------------------- edge score + segment max
template <int H>
__global__ void edge_score_max(const int* __restrict__ src,
                               const int* __restrict__ dst,
                               const float* __restrict__ el,
                               const float* __restrict__ er,
                               float* __restrict__ ebuf,
                               float* __restrict__ m, long EH) {
  long i = (long)blockIdx.x * blockDim.x + threadIdx.x;
  if (i >= EH) return;
  long e = i / H; int h = (int)(i & (H - 1)); // power-of-two H -> shift/and
  float x = el[(long)src[e] * H + h] + er[(long)dst[e] * H + h];
  x = (x > 0.f) ? x : 0.2f * x;               // leaky_relu, slope 0.2
  ebuf[i] = x;
  atomicMaxF(&m[(long)dst[e] * H + h], x);
}

// ------------------------------------------------- exp + segment sum
template <int H>
__global__ void edge_exp_sum(const int* __restrict__ dst,
                             float* __restrict__ ebuf,
                             const float* __restrict__ m,
                             float* __restrict__ denom, long EH) {
  long i = (long)blockIdx.x * blockDim.x + threadIdx.x;
  if (i >= EH) return;
  long e = i / H; int h = (int)(i & (H - 1));
  long di = (long)dst[e] * H + h;
  float a = expf(ebuf[i] - m[di]);
  ebuf[i] = a;
  atomicAdd(&denom[di], a);
}

// ------------------------------------------------- weighted scatter-add
// one thread per (edge, 4-elem chunk): float4 gather of feat[src] +
// 4x global_atomic_add_f32 into out[dst].  Chunk never crosses a head.
template <int F, int D>
__global__ void edge_scatter(const int* __restrict__ src,
                             const int* __restrict__ dst,
                             const float* __restrict__ feat,
                             const float* __restrict__ ebuf,
                             const float* __restrict__ denom,
                             float* __restrict__ out, long nchunks) {
  constexpr int C4 = F / 4;                   // chunks per edge
  constexpr int H  = F / D;
  long i = (long)blockIdx.x * blockDim.x + threadIdx.x;
  if (i >= nchunks) return;
  long e = i / C4; int c = (int)(i & (C4 - 1));
  int  r = c * 4;  int h = r / D;             // all compile-time shifts
  long si = src[e], di = dst[e];
  float alpha = ebuf[e * H + h] / denom[di * H + h];
  float4 f = *(const float4*)(feat + si * F + r);
  float* o = out + di * F + r;
  atomicAdd(o + 0, f.x * alpha);
  atomicAdd(o + 1, f.y * alpha);
  atomicAdd(o + 2, f.z * alpha);
  atomicAdd(o + 3, f.w * alpha);
}

// ------------------------------------------------- bias (+optional ReLU)
template <int F, bool RELU>
__global__ void bias_act(const float* __restrict__ in,
                         const float* __restrict__ bias,
                         float* __restrict__ out, long n4) {
  long i = (long)blockIdx.x * blockDim.x + threadIdx.x;   // over float4 lanes
  if (i >= n4) return;
  int c = (int)(i & (F / 4 - 1));
  float4 v = ((const float4*)in)[i];
  float4 b = ((const float4*)bias)[c];
  v.x += b.x; v.y += b.y; v.z += b.z; v.w += b.w;
  if (RELU) {
    v.x = fmaxf(v.x, 0.f); v.y = fmaxf(v.y, 0.f);
    v.z = fmaxf(v.z, 0.f); v.w = fmaxf(v.w, 0.f);
  }
  ((float4*)out)[i] = v;
}

// ---------------------------------------------------------------- launcher
extern "C" void kernel_launch(void* const* d_in, const int* in_sizes, int n_in,
                              void* d_out, int out_size, void* d_ws, size_t ws_size,
                              hipStream_t stream) {
  const float* h   = (const float*)d_in[0];
  const int*   src = (const int*)d_in[1];
  const int*   dst = (const int*)d_in[2];
  const float* W1  = (const float*)d_in[3];
  const float* al1 = (const float*)d_in[4];
  const float* ar1 = (const float*)d_in[5];
  const float* b1  = (const float*)d_in[6];
  const float* W2  = (const float*)d_in[7];
  const float* al2 = (const float*)d_in[8];
  const float* ar2 = (const float*)d_in[9];
  const float* b2  = (const float*)d_in[10];

  constexpr int F1 = 128, H1 = 4, D1 = 32;   // layer 1: [N,128] -> [N,4,32]
  constexpr int F2 = 64,  H2 = 1, D2 = 64;   // layer 2: [N,128] -> [N,1,64]
  const long N = in_sizes[0] / F1;
  const long E = in_sizes[1];

  // workspace layout (floats), ~80 MB total with aliasing
  float* ws    = (float*)d_ws;
  float* feat1 = ws;                         // N*128
  float* out1  = feat1 + N * F1;             // N*128
  float* feat2 = out1  + N * F1;             // N*64
  float* ebuf  = feat2 + N * F2;             // E*H1 (layer2 reuses, E*H2)
  float* el    = ebuf  + E * H1;             // N*H1 (layer2 reuses)
  float* er    = el    + N * H1;
  float* mbuf  = er    + N * H1;
  float* den   = mbuf  + N * H1;
  float* x2    = feat1;                      // alias: feat1 dead after scatter1
  float* out2  = out1;                       // alias: out1 dead after bias_relu

  auto cdiv = [](long a, long b) { return (unsigned)((a + b - 1) / b); };
  const int T = 256;
  const unsigned mtiles = cdiv(N, 16);

  // ---------------- layer 1 ----------------
  fill_kernel<<<cdiv(N * H1, T), T, 0, stream>>>(mbuf, -INFINITY, N * H1);
  fill_kernel<<<cdiv(N * H1, T), T, 0, stream>>>(den, 0.f, N * H1);
  fill_kernel<<<cdiv(N * F1, T), T, 0, stream>>>(out1, 0.f, N * F1);

  wmma_gemm_f32_k128<F1><<<mtiles, 32 * (F1 / 16), 0, stream>>>(h, W1, feat1, (int)N);
  attn_dots<H1, D1><<<cdiv(N * H1, T), T, 0, stream>>>(feat1, al1, ar1, el, er, N * H1);
  edge_score_max<H1><<<cdiv(E * H1, T), T, 0, stream>>>(src, dst, el, er, ebuf, mbuf, E * H1);
  edge_exp_sum<H1><<<cdiv(E * H1, T), T, 0, stream>>>(dst, ebuf, mbuf, den, E * H1);
  edge_scatter<F1, D1><<<cdiv(E * (F1 / 4), T), T, 0, stream>>>(src, dst, feat1, ebuf, den,
                                                                out1, E * (F1 / 4));
  bias_act<F1, true><<<cdiv(N * F1 / 4, T), T, 0, stream>>>(out1, b1, x2, N * F1 / 4);

  // ---------------- layer 2 ----------------
  fill_kernel<<<cdiv(N * H2, T), T, 0, stream>>>(mbuf, -INFINITY, N * H2);
  fill_kernel<<<cdiv(N * H2, T), T, 0, stream>>>(den, 0.f, N * H2);
  fill_kernel<<<cdiv(N * F2, T), T, 0, stream>>>(out2, 0.f, N * F2);

  wmma_gemm_f32_k128<F2><<<mtiles, 32 * (F2 / 16), 0, stream>>>(x2, W2, feat2, (int)N);
  attn_dots<H2, D2><<<cdiv(N * H2, T), T, 0, stream>>>(feat2, al2, ar2, el, er, N * H2);
  edge_score_max<H2><<<cdiv(E * H2, T), T, 0, stream>>>(src, dst, el, er, ebuf, mbuf, E * H2);
  edge_exp_sum<H2><<<cdiv(E * H2, T), T, 0, stream>>>(dst, ebuf, mbuf, den, E * H2);
  edge_scatter<F2, D2><<<cdiv(E * (F2 / 4), T), T, 0, stream>>>(src, dst, feat2, ebuf, den,
                                                                out2, E * (F2 / 4));
  bias_act<F2, false><<<cdiv(N * F2 / 4, T), T, 0, stream>>>(out2, b2, (float*)d_out, N * F2 / 4);
}